// DCTLayer_54520314856134
// MI455X (gfx1250) — compile-verified
//
#include <hip/hip_runtime.h>

typedef __attribute__((ext_vector_type(2))) float v2f;
typedef __attribute__((ext_vector_type(4))) float v4f;
typedef __attribute__((ext_vector_type(8))) float v8f;

#define LDS_STRIDE 20          // floats per LDS tile row (16 data + 4 pad) -> 80B, 16B aligned
#define TILE_F (16 * LDS_STRIDE)
#define WAVES_PER_BLOCK 8

// Compile-time geometry: x is [16, 3, 1024, 1024] f32.
constexpr int W_C          = 1024;
constexpr int TILES_ROW_LG = 6;    // 1024/16 = 64 tiles per row
constexpr int TILES_IMG_LG = 12;   // 64*64 = 4096 tiles per image plane

// One wave32 processes one 16x16 tile = 2x2 group of 8x8 DCT blocks.
// out = Dblk * X * Dblk^T, computed as U = Dblk * X^T ; out = Dblk * U^T,
// with Dblk = diag(D, D) held in the WMMA A-operand layout.
__launch_bounds__(256)
__global__ void dct8x8_wmma_kernel(const float* __restrict__ x,
                                   const float* __restrict__ dct,
                                   const int* __restrict__ invp,
                                   float* __restrict__ out) {
    __shared__ float sh[WAVES_PER_BLOCK * TILE_F];

    const int tid  = threadIdx.x;
    const int wave = tid >> 5;
    const int lane = tid & 31;
    const int hi   = lane >> 4;   // which half-wave (selects K pair / row half)
    const int lm   = lane & 15;   // m (for A) or n (for B/C/D)

    const int inv = invp[0];

    // ---- Build A operand: Dblk = diag(D, D), 4 chunks of K=4 ----
    // A 16x4 f32 layout: lanes 0-15 hold M=lane {K=0,1 in v0,v1},
    //                    lanes 16-31 hold M=lane-16 {K=2,3 in v0,v1}.
    v2f a[4];
#pragma unroll
    for (int c = 0; c < 4; ++c) {
#pragma unroll
        for (int v = 0; v < 2; ++v) {
            const int k = 4 * c + v + 2 * hi;   // global K column of Dblk
            const int m = lm;                   // row of Dblk
            float val = 0.0f;
            if ((m >> 3) == (k >> 3)) {         // block-diagonal
                const int i = m & 7, j = k & 7;
                val = inv ? dct[j * 8 + i] : dct[i * 8 + j];
            }
            a[c][v] = val;
        }
    }

    // ---- Tile coordinates (all power-of-two: shifts only) ----
    const unsigned tileId = blockIdx.x * WAVES_PER_BLOCK + wave;
    const unsigned img = tileId >> TILES_IMG_LG;                       // plane index
    const unsigned tr  = (tileId >> TILES_ROW_LG) & ((1u << TILES_ROW_LG) - 1);
    const unsigned tc  = tileId & ((1u << TILES_ROW_LG) - 1);
    const size_t base  = ((size_t)img << 20) + ((size_t)tr << 14) + ((size_t)tc << 4);

    float* myLds = sh + wave * TILE_F;

    // ---- Phase 1: coalesced global -> LDS (row-major, padded stride) ----
    {
        const int q  = lane & 3;    // column quad (16B)
        const int rh = lane >> 2;   // row 0..7
        const float* src = x + base + (size_t)rh * W_C + q * 4;
        float*       dst = myLds + rh * LDS_STRIDE + q * 4;
        v4f v0 = *(const v4f*)(src);
        v4f v1 = *(const v4f*)(src + 8 * W_C);        // imm offset 32768B
        *(v4f*)(dst) = v0;
        *(v4f*)(dst + 8 * LDS_STRIDE) = v1;
    }
    asm volatile("" ::: "memory");   // keep LDS phases ordered (HW keeps same-wave LDS in order)

    // ---- Phase 2: U = Dblk * X^T ----
    // B 4x16 layout (mirrors C/D convention): vgpr v, lanes 0-15 -> K=4c+v,
    // lanes 16-31 -> K=4c+v+2; N = lane&15. B(k,n) = X^T(k,n) = X(n,k).
    const float* tcol = myLds + lm * LDS_STRIDE + 2 * hi;
    v8f acc = {};
#pragma unroll
    for (int c = 0; c < 4; ++c) {
        v2f b;
        b[0] = tcol[4 * c + 0];
        b[1] = tcol[4 * c + 1];
        acc = __builtin_amdgcn_wmma_f32_16x16x4_f32(
            false, a[c], false, b, (short)0, acc, false, false);
    }
    asm volatile("" ::: "memory");

    // ---- Phase 3: store U into LDS (C/D layout: vgpr v -> rows v, v+8) ----
    {
        float* urow = myLds + (8 * hi) * LDS_STRIDE + lm;
#pragma unroll
        for (int v = 0; v < 8; ++v) {
            urow[v * LDS_STRIDE] = acc[v];
        }
    }
    asm volatile("" ::: "memory");

    // ---- Phase 4: out = Dblk * U^T ----
    v8f res = {};
#pragma unroll
    for (int c = 0; c < 4; ++c) {
        v2f b;
        b[0] = tcol[4 * c + 0];
        b[1] = tcol[4 * c + 1];
        res = __builtin_amdgcn_wmma_f32_16x16x4_f32(
            false, a[c], false, b, (short)0, res, false, false);
    }

    // ---- Phase 5: store result; one base address + 8 imm-offset stores ----
    {
        float* dst = out + base + (size_t)(8 * hi) * W_C + lm;
#pragma unroll
        for (int v = 0; v < 8; ++v) {
            dst[v * W_C] = res[v];                    // imm offsets v*4096B
        }
    }
}

extern "C" void kernel_launch(void* const* d_in, const int* in_sizes, int n_in,
                              void* d_out, int out_size, void* d_ws, size_t ws_size,
                              hipStream_t stream) {
    const float* x   = (const float*)d_in[0];
    const float* dct = (const float*)d_in[1];
    const int*   inv = (const int*)d_in[2];
    float*       out = (float*)d_out;

    const long n      = (long)in_sizes[0];   // 50,331,648 elements
    const long tiles  = n / 256;             // 16x16 tiles
    const int  blocks = (int)(tiles / WAVES_PER_BLOCK);

    dct8x8_wmma_kernel<<<blocks, 256, 0, stream>>>(x, dct, inv, out);
}